// SelfAttention_87282325389796
// MI455X (gfx1250) — compile-verified
//
#include <hip/hip_runtime.h>

// ---------------------------------------------------------------------------
// Performer linear attention for MI455X (gfx1250).
// f32 WMMA (16x16x4) + async global->LDS staging (ASYNCcnt) + double buffering.
// ---------------------------------------------------------------------------

typedef float v2f __attribute__((ext_vector_type(2)));
typedef float v8f __attribute__((ext_vector_type(8)));

#define B_    4
#define S_    4096
#define DIM_  1024
#define H_    16
#define M_    256
#define DH_   64
#define NR_   (B_ * S_)          // 16384 rows
#define KEPS  1e-3f
#define PHI_SCALE 0.35355339059327373f   // 64^-0.25

__device__ __forceinline__ v8f vzero8() {
  v8f z = {0.f, 0.f, 0.f, 0.f, 0.f, 0.f, 0.f, 0.f};
  return z;
}

// D = A(16x4) * B(4x16) + C, f32 in/out.  Emits v_wmma_f32_16x16x4_f32.
__device__ __forceinline__ v8f wmma4(v2f a, v2f b, v8f c) {
  return __builtin_amdgcn_wmma_f32_16x16x4_f32(false, a, false, b, 0, c,
                                               false, false);
}

// Async copy 16B global -> LDS (GVS mode: saddr base + 32-bit voffset).
// LDS dest = low 32 bits of the flat shared pointer (wave-relative LDS addr).
__device__ __forceinline__ void async_g2l_b128(const void* gbase, int goff,
                                               const void* lptr) {
  unsigned ldsOff = (unsigned)(size_t)lptr;
  asm volatile("global_load_async_to_lds_b128 %0, %1, %2"
               :
               : "v"(ldsOff), "v"(goff),
                 "s"((unsigned long long)(size_t)gbase)
               : "memory");
}
__device__ __forceinline__ void wait_async0() {
  asm volatile("s_wait_asynccnt 0" ::: "memory");
}

// ---------------------------------------------------------------------------
// Kernel 1/6: C[NR,1024] = A[NR,K] * W[K,1024] (+ optional bias)
// Block tile 128x128, BK=16, 8 waves as 4(M) x 2(N); wave tile 32x64.
// Double-buffered async LDS pipeline: tile k+1 loads overlap tile k WMMAs.
// ---------------------------------------------------------------------------
__global__ __launch_bounds__(256)
void gemm_f32(const float* __restrict__ A, const float* __restrict__ W,
              const float* __restrict__ bias, float* __restrict__ C, int K) {
  __shared__ __align__(16) float As[2][128][20];  // stride 20: conflict-free
  __shared__ __align__(16) float Bs[2][16][132];

  const int t    = threadIdx.x;
  const int lane = t & 31, lo = lane & 15, hi = lane >> 4;
  const int w    = t >> 5, wm = w >> 1, wn = w & 1;
  const int rowBase = blockIdx.x * 128;
  const int colBase = blockIdx.y * 128;

  v8f acc[2][4];
#pragma unroll
  for (int i = 0; i < 2; ++i)
#pragma unroll
    for (int j = 0; j < 4; ++j) acc[i][j] = vzero8();

  // Per-thread tile-slice coordinates for the async copies.
  const int ar0 = t >> 2,        ac0 = (t & 3) * 4;          // A: 2 slices
  const int ar1 = (t >> 2) + 64;
  const int br0 = t >> 5,        bc0 = (t & 31) * 4;         // B: 2 slices
  const int br1 = (t >> 5) + 8;

  auto issue = [&](int k0, int buf) {
    async_g2l_b128(A, (int)(((rowBase + ar0) * K + k0 + ac0) * 4),
                   &As[buf][ar0][ac0]);
    async_g2l_b128(A, (int)(((rowBase + ar1) * K + k0 + ac0) * 4),
                   &As[buf][ar1][ac0]);
    async_g2l_b128(W, (int)(((k0 + br0) * DIM_ + colBase + bc0) * 4),
                   &Bs[buf][br0][bc0]);
    async_g2l_b128(W, (int)(((k0 + br1) * DIM_ + colBase + bc0) * 4),
                   &Bs[buf][br1][bc0]);
  };

  issue(0, 0);
  int cur = 0;
  for (int k0 = 0; k0 < K; k0 += 16) {
    wait_async0();          // my slices of tile k0 (and any prefetch) are in LDS
    __syncthreads();        // everyone's slices landed; prev compute finished
    if (k0 + 16 < K) issue(k0 + 16, cur ^ 1);   // overlap with compute below

    const float (*Ab)[20]  = As[cur];
    const float (*Bb)[132] = Bs[cur];
#pragma unroll
    for (int ks = 0; ks < 4; ++ks) {
      const int kk = ks * 4 + 2 * hi;
      v2f a[2], b[4];
#pragma unroll
      for (int mt = 0; mt < 2; ++mt) {
        int mr = wm * 32 + mt * 16 + lo;
        a[mt].x = Ab[mr][kk];
        a[mt].y = Ab[mr][kk + 1];
      }
#pragma unroll
      for (int nt = 0; nt < 4; ++nt) {
        int nc = wn * 64 + nt * 16 + lo;
        b[nt].x = Bb[kk][nc];
        b[nt].y = Bb[kk + 1][nc];
      }
#pragma unroll
      for (int mt = 0; mt < 2; ++mt)
#pragma unroll
        for (int nt = 0; nt < 4; ++nt)
          acc[mt][nt] = wmma4(a[mt], b[nt], acc[mt][nt]);
    }
    cur ^= 1;
  }

#pragma unroll
  for (int mt = 0; mt < 2; ++mt)
#pragma unroll
    for (int nt = 0; nt < 4; ++nt)
#pragma unroll
      for (int r = 0; r < 8; ++r) {
        int row = rowBase + wm * 32 + mt * 16 + r + 8 * hi;
        int col = colBase + wn * 64 + nt * 16 + lo;
        float v = acc[mt][nt][r];
        if (bias) v += bias[col];
        C[(size_t)row * DIM_ + col] = v;
      }
}

// ---------------------------------------------------------------------------
// Kernel 3: QP[bh,s,m] = relu(scale * Q[b,s,h*64+d] * proj[m,d]) + eps
// Per (b,h) GEMM [128 x 64] x [64 x 128], BK=32 (2 iters), waves 4x2.
// Q tile staged with async copies; proj tile needs a transpose (plain stores).
// ---------------------------------------------------------------------------
__global__ __launch_bounds__(256)
void phi_kernel(const float* __restrict__ Q, const float* __restrict__ proj,
                float* __restrict__ QP) {
  __shared__ __align__(16) float As[128][36];
  __shared__ __align__(16) float Bs[32][132];

  const int t    = threadIdx.x;
  const int lane = t & 31, lo = lane & 15, hi = lane >> 4;
  const int w    = t >> 5, wm = w >> 1, wn = w & 1;
  const int sBase   = blockIdx.x * 128;
  const int colBase = blockIdx.y * 128;
  const int bh = blockIdx.z;
  const int b = bh >> 4, h = bh & 15;
  const float* Qb  = Q + (size_t)b * S_ * DIM_ + h * DH_;
  float*       QPb = QP + (size_t)bh * S_ * M_;

  v8f acc[2][4];
#pragma unroll
  for (int i = 0; i < 2; ++i)
#pragma unroll
    for (int j = 0; j < 4; ++j) acc[i][j] = vzero8();

  for (int kb = 0; kb < DH_; kb += 32) {
    __syncthreads();
#pragma unroll
    for (int p = 0; p < 4; ++p) {          // Q tile 128x32 via async copies
      int idx = p * 256 + t;
      int r = idx >> 3, c4 = (idx & 7) * 4;
      async_g2l_b128(Qb, (int)(((sBase + r) * DIM_ + kb + c4) * 4),
                     &As[r][c4]);
    }
#pragma unroll
    for (int p = 0; p < 4; ++p) {          // projT tile 32x128 (transposed)
      int idx = p * 256 + t;
      int n = idx >> 3, k4 = idx & 7;
      float4 v = *reinterpret_cast<const float4*>(
          &proj[(size_t)(colBase + n) * DH_ + kb + k4 * 4]);
      Bs[k4 * 4 + 0][n] = v.x;
      Bs[k4 * 4 + 1][n] = v.y;
      Bs[k4 * 4 + 2][n] = v.z;
      Bs[k4 * 4 + 3][n] = v.w;
    }
    wait_async0();
    __syncthreads();
#pragma unroll
    for (int ks = 0; ks < 8; ++ks) {
      const int kk = ks * 4 + 2 * hi;
      v2f a[2], b[4];
#pragma unroll
      for (int mt = 0; mt < 2; ++mt) {
        int mr = wm * 32 + mt * 16 + lo;
        a[mt].x = As[mr][kk];
        a[mt].y = As[mr][kk + 1];
      }
#pragma unroll
      for (int nt = 0; nt < 4; ++nt) {
        int nc = wn * 64 + nt * 16 + lo;
        b[nt].x = Bs[kk][nc];
        b[nt].y = Bs[kk + 1][nc];
      }
#pragma unroll
      for (int mt = 0; mt < 2; ++mt)
#pragma unroll
        for (int nt = 0; nt < 4; ++nt)
          acc[mt][nt] = wmma4(a[mt], b[nt], acc[mt][nt]);
    }
  }

#pragma unroll
  for (int mt = 0; mt < 2; ++mt)
#pragma unroll
    for (int nt = 0; nt < 4; ++nt)
#pragma unroll
      for (int r = 0; r < 8; ++r) {
        int row = sBase + wm * 32 + mt * 16 + r + 8 * hi;
        int col = colBase + wn * 64 + nt * 16 + lo;
        float val = fmaxf(acc[mt][nt][r] * PHI_SCALE, 0.f) + KEPS;
        QPb[(size_t)row * M_ + col] = val;
      }
}

// ---------------------------------------------------------------------------
// Kernel 4: KVext[bh,m,0:64] += sum_s kp[s,m]*v[s,d] ; KVext[bh,m,64] += z[m]
// Split-K over S (8 chunks of 512); ones-column trick computes z in-GEMM.
// Each wave: 32 m-rows x 80 cols (2x5 tiles). f32 atomic accumulation.
// ---------------------------------------------------------------------------
__global__ __launch_bounds__(256)
void kv_kernel(const float* __restrict__ QP, const float* __restrict__ V,
               float* __restrict__ KVext) {
  __shared__ __align__(16) float KPs[32][260];  // [s][m]
  __shared__ __align__(16) float Vs[32][84];    // [s][d | z-cols 64..79]

  const int t    = threadIdx.x;
  const int lane = t & 31, lo = lane & 15, hi = lane >> 4;
  const int w    = t >> 5;                 // wave -> m strip of 32
  const int chunk = blockIdx.x;            // 0..7
  const int bh = blockIdx.z;
  const int b = bh >> 4, h = bh & 15;
  const float* QPb = QP + (size_t)bh * S_ * M_;
  const float* Vb  = V + (size_t)b * S_ * DIM_ + h * DH_;
  float*       KVb = KVext + (size_t)bh * M_ * 80;

  v8f acc[2][5];
#pragma unroll
  for (int i = 0; i < 2; ++i)
#pragma unroll
    for (int j = 0; j < 5; ++j) acc[i][j] = vzero8();

  // Fill z-columns once: col 64 = 1, cols 65..79 = 0 (async loads never touch them)
#pragma unroll
  for (int p = 0; p < 2; ++p) {
    int idx = p * 256 + t;
    int r = idx >> 4, c = idx & 15;
    Vs[r][64 + c] = (c == 0) ? 1.0f : 0.0f;
  }

  for (int st = 0; st < 512; st += 32) {
    const int sG = chunk * 512 + st;
    __syncthreads();
#pragma unroll
    for (int p = 0; p < 8; ++p) {          // KP tile 32x256 via async
      int idx = p * 256 + t;
      int r = idx >> 6, c4 = (idx & 63) * 4;
      async_g2l_b128(QPb, (int)(((sG + r) * M_ + c4) * 4), &KPs[r][c4]);
    }
#pragma unroll
    for (int p = 0; p < 2; ++p) {          // V tile 32x64 via async
      int idx = p * 256 + t;
      int r = idx >> 4, c4 = (idx & 15) * 4;
      async_g2l_b128(Vb, (int)(((sG + r) * DIM_ + c4) * 4), &Vs[r][c4]);
    }
    wait_async0();
    __syncthreads();
#pragma unroll
    for (int ks = 0; ks < 8; ++ks) {
      const int kk = ks * 4 + 2 * hi;
      v2f a[2], b[5];
#pragma unroll
      for (int mt = 0; mt < 2; ++mt) {
        int mr = w * 32 + mt * 16 + lo;    // A = KP^T : A[m][s] = KPs[s][m]
        a[mt].x = KPs[kk][mr];
        a[mt].y = KPs[kk + 1][mr];
      }
#pragma unroll
      for (int nt = 0; nt < 5; ++nt) {
        int nc = nt * 16 + lo;
        b[nt].x = Vs[kk][nc];
        b[nt].y = Vs[kk + 1][nc];
      }
#pragma unroll
      for (int mt = 0; mt < 2; ++mt)
#pragma unroll
        for (int nt = 0; nt < 5; ++nt)
          acc[mt][nt] = wmma4(a[mt], b[nt], acc[mt][nt]);
    }
  }

#pragma unroll
  for (int mt = 0; mt < 2; ++mt)
#pragma unroll
    for (int nt = 0; nt < 5; ++nt)
#pragma unroll
      for (int r = 0; r < 8; ++r) {
        int m = w * 32 + mt * 16 + r + 8 * hi;
        int n = nt * 16 + lo;
        if (n <= 64) atomicAdd(&KVb[(size_t)m * 80 + n], acc[mt][nt][r]);
      }
}

// ---------------------------------------------------------------------------
// Kernel 5: num/den + normalize.
// Attn[b,s,h*64+d] = (QP[s,:] @ KV[:,d]) / (QP[s,:] @ z)   (den = col 64)
// Each wave: 16 rows x 80 cols (5 tiles); K=256 in BK=32 steps.
// ---------------------------------------------------------------------------
__global__ __launch_bounds__(256)
void num_kernel(const float* __restrict__ QP, const float* __restrict__ KVext,
                float* __restrict__ Attn) {
  __shared__ __align__(16) float QPs[128][36];
  __shared__ __align__(16) float KVs[32][84];

  const int t    = threadIdx.x;
  const int lane = t & 31, lo = lane & 15, hi = lane >> 4;
  const int w    = t >> 5;
  const int sBase = blockIdx.x * 128;
  const int bh = blockIdx.z;
  const int b = bh >> 4, h = bh & 15;
  const float* QPb = QP + (size_t)bh * S_ * M_;
  const float* KVb = KVext + (size_t)bh * M_ * 80;
  float*       Ab  = Attn + (size_t)b * S_ * DIM_ + h * DH_;

  v8f acc[5];
#pragma unroll
  for (int j = 0; j < 5; ++j) acc[j] = vzero8();

  for (int kb = 0; kb < M_; kb += 32) {
    __syncthreads();
#pragma unroll
    for (int p = 0; p < 4; ++p) {          // QP tile 128x32 via async
      int idx = p * 256 + t;
      int r = idx >> 3, c4 = (idx & 7) * 4;
      async_g2l_b128(QPb, (int)(((sBase + r) * M_ + kb + c4) * 4),
                     &QPs[r][c4]);
    }
    for (int i = t; i < 32 * 20; i += 256) {  // KV tile 32x80 via async
      int r = i / 20, c4 = (i % 20) * 4;
      async_g2l_b128(KVb, (int)(((kb + r) * 80 + c4) * 4), &KVs[r][c4]);
    }
    wait_async0();
    __syncthreads();
#pragma unroll
    for (int ks = 0; ks < 8; ++ks) {
      const int kk = ks * 4 + 2 * hi;
      const int mr = w * 16 + lo;
      v2f a;
      a.x = QPs[mr][kk];
      a.y = QPs[mr][kk + 1];
#pragma unroll
      for (int nt = 0; nt < 5; ++nt) {
        v2f bb;
        int nc = nt * 16 + lo;
        bb.x = KVs[kk][nc];
        bb.y = KVs[kk + 1][nc];
        acc[nt] = wmma4(a, bb, acc[nt]);
      }
    }
  }

  // den for row m sits at tile 4, N=64 -> lane 0 (m=r) / lane 16 (m=r+8).
#pragma unroll
  for (int r = 0; r < 8; ++r) {
    float den = __shfl(acc[4][r], lane & 16, 32);
    float inv = 1.0f / den;
    int row = sBase + w * 16 + r + 8 * hi;
#pragma unroll
    for (int nt = 0; nt < 4; ++nt)
      Ab[(size_t)row * DIM_ + nt * 16 + lo] = acc[nt][r] * inv;
  }
}

// ---------------------------------------------------------------------------
// Launch: inputs = x, Wq, Wv, Wo, bo, proj (all f32). Output f32 [B,S,DIM].
// ws layout: [q/attn 64MB][v 64MB][qp 64MB][kv_ext 5.25MB]  (~206 MB)
// ---------------------------------------------------------------------------
extern "C" void kernel_launch(void* const* d_in, const int* in_sizes, int n_in,
                              void* d_out, int out_size, void* d_ws,
                              size_t ws_size, hipStream_t stream) {
  (void)in_sizes; (void)n_in; (void)out_size; (void)ws_size;
  const float* x    = (const float*)d_in[0];
  const float* Wq   = (const float*)d_in[1];
  const float* Wv   = (const float*)d_in[2];
  const float* Wo   = (const float*)d_in[3];
  const float* bo   = (const float*)d_in[4];
  const float* proj = (const float*)d_in[5];
  float* out = (float*)d_out;

  char* ws = (char*)d_ws;
  const size_t szQ = (size_t)NR_ * DIM_ * sizeof(float);  // 64 MB
  float* q     = (float*)(ws);
  float* v     = (float*)(ws + szQ);
  float* qp    = (float*)(ws + 2 * szQ);
  float* kvext = (float*)(ws + 3 * szQ);
  float* attn  = q;  // q is dead after phi; reuse for attention output

  dim3 blk(256);
  dim3 gGemm(NR_ / 128, DIM_ / 128);           // 128 x 8
  gemm_f32<<<gGemm, blk, 0, stream>>>(x, Wq, (const float*)nullptr, q, DIM_);
  gemm_f32<<<gGemm, blk, 0, stream>>>(x, Wv, (const float*)nullptr, v, DIM_);

  dim3 gPhi(S_ / 128, M_ / 128, B_ * H_);      // 32 x 2 x 64
  phi_kernel<<<gPhi, blk, 0, stream>>>(q, proj, qp);

  (void)hipMemsetAsync(kvext, 0, (size_t)B_ * H_ * M_ * 80 * sizeof(float),
                       stream);
  dim3 gKV(8, 1, B_ * H_);                     // split-K over S
  kv_kernel<<<gKV, blk, 0, stream>>>(qp, v, kvext);

  dim3 gNum(S_ / 128, 1, B_ * H_);
  num_kernel<<<gNum, blk, 0, stream>>>(qp, kvext, attn);

  gemm_f32<<<gGemm, blk, 0, stream>>>(attn, Wo, bo, out, DIM_);
}